// RQ_k_means_46600395162147
// MI455X (gfx1250) — compile-verified
//
#include <hip/hip_runtime.h>

typedef float v2f __attribute__((ext_vector_type(2)));
typedef float v8f __attribute__((ext_vector_type(8)));

#define WGSZ   256
#define WAVES  8
#define ROWS   32      // vectors (rows) per workgroup = two 16-row WMMA tiles
#define DIM    64      // vector dimension
#define KCB    1024    // codes per stage
#define NSTAGE 4
#define COLS_PER_WAVE (KCB / WAVES)           // 128
#define TILES_PER_WAVE (COLS_PER_WAVE / 16)   // 8
#define NVEC   32768   // 8*4096
#define LOSS_SCALE (1.25f / (float)(NVEC * DIM))

// ---------------------------------------------------------------------------
// Pre-kernel: c2[s*K + k] = ||codebook[s][k]||^2   (4096 entries, 16 KB)
// ---------------------------------------------------------------------------
__global__ __launch_bounds__(256) void rq_c2_kernel(const float* __restrict__ cb,
                                                    float* __restrict__ c2) {
    int idx = blockIdx.x * blockDim.x + threadIdx.x;   // 0..4095
    const float4* p = (const float4*)(cb + (size_t)idx * DIM);
    float s = 0.0f;
#pragma unroll
    for (int q = 0; q < DIM / 4; ++q) {
        float4 v = p[q];
        s += v.x * v.x + v.y * v.y + v.z * v.z + v.w * v.w;
    }
    c2[idx] = s;
}

// ---------------------------------------------------------------------------
// Main kernel: residual VQ over a 32-row tile (2 WMMA row-tiles per wave).
// Each codebook B load feeds two independent v_wmma_f32_16x16x4_f32 chains.
// ---------------------------------------------------------------------------
__global__ __launch_bounds__(WGSZ) void rq_vq_kernel(const float* __restrict__ z,
                                                     const float* __restrict__ cb,
                                                     const float* __restrict__ c2g,
                                                     float* __restrict__ zq,
                                                     float* __restrict__ loss_out) {
    __shared__ __align__(16) float lds_resid[ROWS][DIM];   // live residual tile (8 KB)
    __shared__ __align__(16) float lds_c2[NSTAGE * KCB];   // code norms (16 KB)
    __shared__ float lds_wmin[WAVES][ROWS];
    __shared__ int   lds_widx[WAVES][ROWS];
    __shared__ int   lds_best[ROWS];
    __shared__ float lds_loss[WAVES];

    const int tid  = threadIdx.x;
    const int lane = tid & 31;
    const int wave = tid >> 5;
    const int wg   = blockIdx.x;

    // Stage c2 into LDS (1024 float4 loads spread over 256 threads).
    {
        const float4* src = (const float4*)c2g;
        float4* dst = (float4*)lds_c2;
#pragma unroll
        for (int j = 0; j < 4; ++j) dst[tid * 4 + j] = src[tid * 4 + j];
    }

    // Each thread owns 4 contiguous elements of row `row` and of row `row+16`.
    const int row  = tid >> 4;            // 0..15
    const int col4 = (tid & 15) * 4;      // 0,4,...,60
    const size_t gbase0 = ((size_t)wg * ROWS + row) * DIM + col4;
    const size_t gbase1 = gbase0 + (size_t)16 * DIM;

    {
        float4 zv0 = *(const float4*)(z + gbase0);
        float4 zv1 = *(const float4*)(z + gbase1);
        *(float4*)&lds_resid[row][col4]      = zv0;
        *(float4*)&lds_resid[row + 16][col4] = zv1;
    }

    float4 recon0 = make_float4(0.f, 0.f, 0.f, 0.f);
    float4 recon1 = make_float4(0.f, 0.f, 0.f, 0.f);
    float  lossAcc = 0.0f;

    // WMMA operand geometry for this lane.
    const int halfk     = (lane >> 4) * 2;   // K sub-offset: 0 (lanes 0-15) / 2 (16-31)
    const int arow      = lane & 15;         // A row (M) for this lane
    const int colInTile = lane & 15;         // B/D column (N) within a 16-wide tile

    __syncthreads();

    for (int s = 0; s < NSTAGE; ++s) {
        const float* cbs = cb + (size_t)s * KCB * DIM;

        // Hoist both A operands (shared across all 8 column tiles):
        // A layout: lane<16 holds (K=4k, 4k+1), lane>=16 holds (K=4k+2, 4k+3), M=lane&15.
        v2f A0[16], A1[16];
        {
            const float2* ap0 = (const float2*)&lds_resid[arow][halfk];
            const float2* ap1 = (const float2*)&lds_resid[arow + 16][halfk];
#pragma unroll
            for (int k = 0; k < 16; ++k) {
                float2 a0 = ap0[k * 2];
                float2 a1 = ap1[k * 2];
                A0[k].x = a0.x; A0[k].y = a0.y;
                A1[k].x = a1.x; A1[k].y = a1.y;
            }
        }

        float minv0[8], minv1[8];
        int   mini0[8], mini1[8];
#pragma unroll
        for (int r = 0; r < 8; ++r) {
            minv0[r] = 3.4e38f; mini0[r] = 0;
            minv1[r] = 3.4e38f; mini1[r] = 0;
        }

        for (int t = 0; t < TILES_PER_WAVE; ++t) {
            const int col = wave * COLS_PER_WAVE + t * 16 + colInTile;
            // B layout mirrors A: lane<16 -> (K=4k,4k+1) of column `col`, etc.
            const float2* bp = (const float2*)(cbs + (size_t)col * DIM + halfk);

            v8f acc0 = {};
            v8f acc1 = {};
#pragma unroll
            for (int k = 0; k < 16; ++k) {
                float2 b = bp[k * 2];
                v2f B; B.x = b.x; B.y = b.y;
                // Two independent accumulator chains per B load: 2x reuse,
                // no WMMA->WMMA RAW between consecutive ops.
                acc0 = __builtin_amdgcn_wmma_f32_16x16x4_f32(
                    false, A0[k], false, B, (short)0, acc0, false, false);
                acc1 = __builtin_amdgcn_wmma_f32_16x16x4_f32(
                    false, A1[k], false, B, (short)0, acc1, false, false);
            }

            const float c2v = lds_c2[s * KCB + col];
#pragma unroll
            for (int r = 0; r < 8; ++r) {
                // dist = r2 + c2 - 2*dot ; r2 is row-constant -> drop for argmin.
                float s0 = c2v - 2.0f * acc0[r];
                float s1 = c2v - 2.0f * acc1[r];
                if (s0 < minv0[r]) { minv0[r] = s0; mini0[r] = col; }
                if (s1 < minv1[r]) { minv1[r] = s1; mini1[r] = col; }
            }
        }

        // Intra-wave argmin across the 16 lanes of each half (row half <-> lane half).
#pragma unroll
        for (int m = 8; m >= 1; m >>= 1) {
#pragma unroll
            for (int r = 0; r < 8; ++r) {
                float ov = __shfl_xor(minv0[r], m, 32);
                int   oi = __shfl_xor(mini0[r], m, 32);
                if (ov < minv0[r] || (ov == minv0[r] && oi < mini0[r])) {
                    minv0[r] = ov; mini0[r] = oi;
                }
                float ov1 = __shfl_xor(minv1[r], m, 32);
                int   oi1 = __shfl_xor(mini1[r], m, 32);
                if (ov1 < minv1[r] || (ov1 == minv1[r] && oi1 < mini1[r])) {
                    minv1[r] = ov1; mini1[r] = oi1;
                }
            }
        }
        if ((lane & 15) == 0) {
            const int rbase = (lane >> 4) * 8;   // rows 0-7 from lane 0, 8-15 from lane 16
#pragma unroll
            for (int r = 0; r < 8; ++r) {
                lds_wmin[wave][rbase + r]      = minv0[r];
                lds_widx[wave][rbase + r]      = mini0[r];
                lds_wmin[wave][16 + rbase + r] = minv1[r];
                lds_widx[wave][16 + rbase + r] = mini1[r];
            }
        }
        __syncthreads();

        // Cross-wave argmin (32 rows x 8 wave candidates).
        if (tid < ROWS) {
            float bv = lds_wmin[0][tid];
            int   bi = lds_widx[0][tid];
#pragma unroll
            for (int w = 1; w < WAVES; ++w) {
                float v = lds_wmin[w][tid];
                int   i = lds_widx[w][tid];
                if (v < bv || (v == bv && i < bi)) { bv = v; bi = i; }
            }
            lds_best[tid] = bi;
        }
        __syncthreads();

        // Quantize + update residual / reconstruction / loss for both owned rows.
        {
            const int b0 = lds_best[row];
            const int b1 = lds_best[row + 16];
            float4 q0 = *(const float4*)(cbs + (size_t)b0 * DIM + col4);
            float4 q1 = *(const float4*)(cbs + (size_t)b1 * DIM + col4);

            float4 r0 = *(const float4*)&lds_resid[row][col4];
            float4 r1 = *(const float4*)&lds_resid[row + 16][col4];
            r0.x -= q0.x; r0.y -= q0.y; r0.z -= q0.z; r0.w -= q0.w;
            r1.x -= q1.x; r1.y -= q1.y; r1.z -= q1.z; r1.w -= q1.w;
            *(float4*)&lds_resid[row][col4]      = r0;
            *(float4*)&lds_resid[row + 16][col4] = r1;

            recon0.x += q0.x; recon0.y += q0.y; recon0.z += q0.z; recon0.w += q0.w;
            recon1.x += q1.x; recon1.y += q1.y; recon1.z += q1.z; recon1.w += q1.w;

            // (q - r_old)^2 == r_new^2 ; embedding + 0.25*commit folded into LOSS_SCALE.
            lossAcc += r0.x * r0.x + r0.y * r0.y + r0.z * r0.z + r0.w * r0.w;
            lossAcc += r1.x * r1.x + r1.y * r1.y + r1.z * r1.z + r1.w * r1.w;
        }
        __syncthreads();
    }

    // z_q = reconstructed (straight-through value).
    *(float4*)(zq + gbase0) = recon0;
    *(float4*)(zq + gbase1) = recon1;

    // Loss reduction: wave shuffle -> LDS -> one atomic per WG.
#pragma unroll
    for (int m = 16; m >= 1; m >>= 1) lossAcc += __shfl_xor(lossAcc, m, 32);
    if (lane == 0) lds_loss[wave] = lossAcc;
    __syncthreads();
    if (tid == 0) {
        float t = 0.0f;
#pragma unroll
        for (int w = 0; w < WAVES; ++w) t += lds_loss[w];
        atomicAdd(loss_out, t * LOSS_SCALE);
    }
}

// ---------------------------------------------------------------------------
// Host launch
// ---------------------------------------------------------------------------
extern "C" void kernel_launch(void* const* d_in, const int* in_sizes, int n_in,
                              void* d_out, int out_size, void* d_ws, size_t ws_size,
                              hipStream_t stream) {
    (void)in_sizes; (void)n_in; (void)out_size; (void)ws_size;
    const float* z  = (const float*)d_in[0];   // [8,4096,64] f32
    const float* cb = (const float*)d_in[1];   // [4,1024,64] f32
    float* zq   = (float*)d_out;               // [8,4096,64] f32
    float* loss = zq + (size_t)NVEC * DIM;     // scalar, appended
    float* c2   = (float*)d_ws;                // 16 KB scratch

    hipMemsetAsync(loss, 0, sizeof(float), stream);
    rq_c2_kernel<<<(NSTAGE * KCB) / 256, 256, 0, stream>>>(cb, c2);
    rq_vq_kernel<<<NVEC / ROWS, WGSZ, 0, stream>>>(z, cb, c2, zq, loss);
}